// YOLOV1Loss_16329465659932
// MI455X (gfx1250) — compile-verified
//
#include <hip/hip_runtime.h>
#include <hip/hip_bf16.h>
#include <cstdint>

// ---- problem constants (from reference) ------------------------------------
#define S_DIM    14
#define CH       30                      // 5*B + C = 10 + 20
#define BATCH_N  4096
#define CELLS    (BATCH_N * S_DIM * S_DIM)   // 802816
#define TILE_CELLS  256
#define TILE_FLOATS (TILE_CELLS * CH)        // 7680 floats
#define TILE_BYTES  (TILE_FLOATS * 4)        // 30720 B per tensor per tile
#define NTILES   (CELLS / TILE_CELLS)        // 3136 exactly (no tail)
#define BLOCK    256
#define GRID1    1024
#define NB128    7                           // 7*256*16B = 28672 B via b128
#define TAIL_OFF (NB128 * BLOCK * 16)        // remaining 2048 B via one b64/thread

typedef __attribute__((ext_vector_type(2))) float v2f;
typedef __attribute__((ext_vector_type(8))) float v8f;

// Generic pointer to a __shared__ object is {LDS_aperture_hi32, lds_offset} —
// low 32 bits are the LDS byte address the async-LDS instructions want.
__device__ __forceinline__ unsigned lds_off(const void* p) {
  return (unsigned)(size_t)p;
}

// gfx1250 async copies global -> LDS (ASYNCcnt-tracked, in-order completion).
__device__ __forceinline__ void async_copy_b128(unsigned lds, unsigned long long gaddr) {
  asm volatile("global_load_async_to_lds_b128 %0, %1, off"
               :: "v"(lds), "v"(gaddr) : "memory");
}
__device__ __forceinline__ void async_copy_b64(unsigned lds, unsigned long long gaddr) {
  asm volatile("global_load_async_to_lds_b64 %0, %1, off"
               :: "v"(lds), "v"(gaddr) : "memory");
}
// wait until ASYNCcnt <= n (immediate)
#define WAIT_ASYNC(n) asm volatile("s_wait_asynccnt " #n ::: "memory")

// Issue the full async copy of one tile (16 instructions: 8 per tensor).
__device__ __forceinline__ void issue_tile_copy(unsigned ldsi, unsigned ldst,
                                                const float* inp, const float* tgt,
                                                int tile, int tid) {
  const unsigned long long gin =
      (unsigned long long)(const void*)inp + (unsigned long long)tile * TILE_BYTES;
  const unsigned long long gtg =
      (unsigned long long)(const void*)tgt + (unsigned long long)tile * TILE_BYTES;
  #pragma unroll
  for (int j = 0; j < NB128; ++j) {
    const unsigned e = (unsigned)(j * BLOCK + tid) * 16u;   // byte offset
    async_copy_b128(ldsi + e, gin + e);
    async_copy_b128(ldst + e, gtg + e);
  }
  const unsigned e = (unsigned)TAIL_OFF + (unsigned)tid * 8u;
  async_copy_b64(ldsi + e, gin + e);
  async_copy_b64(ldst + e, gtg + e);
}

// ---- pass 1: double-buffered async-streaming tile reduction ----------------
__global__ __launch_bounds__(BLOCK)
void yolo_partial(const float* __restrict__ inp, const float* __restrict__ tgt,
                  float* __restrict__ partials)
{
  __shared__ alignas(16) float s_in[2][TILE_FLOATS];   // 2 x 30720 B
  __shared__ alignas(16) float s_tg[2][TILE_FLOATS];   // 2 x 30720 B  (~120 KB total)
  __shared__ float s_wsum[BLOCK / 32];

  const int tid = threadIdx.x;
  const unsigned ldsi0 = lds_off(&s_in[0][0]), ldsi1 = lds_off(&s_in[1][0]);
  const unsigned ldst0 = lds_off(&s_tg[0][0]), ldst1 = lds_off(&s_tg[1][0]);

  const int stride = gridDim.x;
  int tile = blockIdx.x;
  int buf  = 0;

  if (tile < NTILES)
    issue_tile_copy(ldsi0, ldst0, inp, tgt, tile, tid);

  float acc = 0.f;
  for (; tile < NTILES; tile += stride, buf ^= 1) {
    const int next = tile + stride;
    if (next < NTILES) {
      // Prefetch next tile into the other buffer (protected by the barrier at
      // the end of the previous iteration), then wait only for THIS tile's 16
      // in-order async ops: <=16 outstanding means current tile has landed.
      issue_tile_copy(buf ? ldsi0 : ldsi1, buf ? ldst0 : ldst1, inp, tgt, next, tid);
      WAIT_ASYNC(16);
    } else {
      WAIT_ASYNC(0);
    }
    __syncthreads();   // all waves' data for `buf` visible

    // One thread = one cell; 30*t mod 64 -> 32 distinct banks per wave.
    const float* p = &s_in[buf][tid * CH];
    const float* t = &s_tg[buf][tid * CH];

    const float t0 = t[0], t1 = t[1];
    const bool obj0 = (t0 == 1.0f);
    const bool obj1 = (t1 == 1.0f);
    const float f0 = obj0 ? 1.f : 0.f;
    const float f1 = (obj1 && !obj0) ? 1.f : 0.f;
    const bool  has = obj0 || obj1;

    const float d0 = p[0] - t0, d1 = p[1] - t1;
    const float sqc0 = d0 * d0, sqc1 = d1 * d1;

    float boxterm;
    {   // box 0: ch 2..5 (x,y,w,h)
      const float dx = p[2] - t[2], dy = p[3] - t[3];
      const float dw = sqrtf(p[4]) - sqrtf(t[4]);
      const float dh = sqrtf(p[5]) - sqrtf(t[5]);
      boxterm = f0 * (dx * dx + dy * dy + dw * dw + dh * dh);
    }
    {   // box 1: ch 6..9
      const float dx = p[6] - t[6], dy = p[7] - t[7];
      const float dw = sqrtf(p[8]) - sqrtf(t[8]);
      const float dh = sqrtf(p[9]) - sqrtf(t[9]);
      boxterm += f1 * (dx * dx + dy * dy + dw * dw + dh * dh);
    }
    float cls = 0.f;
    #pragma unroll
    for (int c = 10; c < CH; ++c) { const float d = p[c] - t[c]; cls += d * d; }

    const float objc  = f0 * sqc0 + f1 * sqc1;
    const float noobj = has ? 0.f : (sqc0 + sqc1);
    acc += 5.0f * boxterm + objc + 0.5f * noobj + (has ? cls : 0.f);

    __syncthreads();   // all waves done reading `buf` before it is refilled
  }

  // wave32 tree reduce, then 8 wave sums -> block partial
  #pragma unroll
  for (int off = 16; off > 0; off >>= 1) acc += __shfl_down(acc, off, 32);
  if ((tid & 31) == 0) s_wsum[tid >> 5] = acc;
  __syncthreads();
  if (tid == 0) {
    float b = 0.f;
    #pragma unroll
    for (int w = 0; w < BLOCK / 32; ++w) b += s_wsum[w];
    partials[blockIdx.x] = b;
  }
}

// ---- pass 2: one full wave; WMMA collapses the 32 lane sums ----------------
__global__ __launch_bounds__(32)
void yolo_final(const float* __restrict__ partials, int npart, float* __restrict__ out)
{
  const int lane = threadIdx.x;
  float s = 0.f;
  for (int i = lane; i < npart; i += 32) s += partials[i];

  // A (16x4 f32, 2 VGPRs/lane) = lane sums in slot0, 0 in slot1; B = ones.
  // Each A slot holds exactly one lane value (or 0) and each D column sums all
  // of A, so sum(D over 8 VGPRs x 32 lanes) == 16 * sum(lane values),
  // independent of the exact VGPR<->element mapping.
  v2f a; a.x = s;   a.y = 0.f;
  v2f b; b.x = 1.f; b.y = 1.f;
  v8f c = {};
  v8f d = __builtin_amdgcn_wmma_f32_16x16x4_f32(
      /*neg_a=*/false, a, /*neg_b=*/false, b,
      /*c_mod=*/(short)0, c, /*reuse_a=*/false, /*reuse_b=*/false);

  float q = d[0] + d[1] + d[2] + d[3] + d[4] + d[5] + d[6] + d[7];
  #pragma unroll
  for (int off = 16; off > 0; off >>= 1) q += __shfl_down(q, off, 32);

  if (lane == 0) out[0] = (q * (1.0f / 16.0f)) * (1.0f / (float)BATCH_N);
}

extern "C" void kernel_launch(void* const* d_in, const int* in_sizes, int n_in,
                              void* d_out, int out_size, void* d_ws, size_t ws_size,
                              hipStream_t stream) {
  const float* inp = (const float*)d_in[0];
  const float* tgt = (const float*)d_in[1];
  float* partials  = (float*)d_ws;

  int nblocks = GRID1;
  if (ws_size < (size_t)GRID1 * sizeof(float)) {
    nblocks = (int)(ws_size / sizeof(float));
    if (nblocks < 1) nblocks = 1;   // degenerate scratch; still deterministic
  }

  yolo_partial<<<nblocks, BLOCK, 0, stream>>>(inp, tgt, partials);
  yolo_final<<<1, 32, 0, stream>>>(partials, nblocks, (float*)d_out);
}